// AggregationLayer_50105088475767
// MI455X (gfx1250) — compile-verified
//
#include <hip/hip_runtime.h>
#include <math.h>

typedef float v2f __attribute__((ext_vector_type(2)));
typedef float v4f __attribute__((ext_vector_type(4)));
typedef float v8f __attribute__((ext_vector_type(8)));

#define HH 480
#define WW 480
#define P  (HH * WW)          // 230400 pixels
#define NINST 64
#define NCOL  80              // 64 data columns + tile4 (col 64 = ones -> mask_size)
#define GEMM_BLOCKS 225
#define CHUNK (P / GEMM_BLOCKS)   // 1024 pixels per block, 128 iterations of 8

// ---------------------------------------------------------------------------
// Kernel 1: S_partial[blk][n][col] = sum_{p in chunk(blk)} mask[n,p] * B[p,col]
// via V_WMMA_F32_16X16X4_F32.  col = b*8+c maps to quat(0-3)/scales(4-6)/z(7);
// column tile 4 is a register-generated "ones" column -> mask_size.
//
// b128 trick: A lanes (M=mr, half h -> K in {2h,2h+1}) and B lanes (N=mr,
// half h -> K in {2h,2h+1}) share the same K-per-lane structure, and the
// pixel->K assignment of a reduction is arbitrary if A and B agree.  Each
// lane loads pixels base+4h+{0..3} as ONE b128; {e0,e1} feed WMMA step A
// (pixels {0,1,4,5}) and {e2,e3} feed step B (pixels {2,3,6,7}).
// ---------------------------------------------------------------------------
__global__ __launch_bounds__(128)
void gemm_partial_kernel(const float* __restrict__ mask,
                         const float* __restrict__ quat,
                         const float* __restrict__ scales,
                         const float* __restrict__ z,
                         float* __restrict__ partials)
{
    const int wave = threadIdx.x >> 5;   // 0..3 : instance row tile
    const int lane = threadIdx.x & 31;
    const int half = lane >> 4;          // 0/1 : K half per WMMA layout
    const int mr   = lane & 15;          // row/col within tile
    const int m0   = wave * 16;
    const int k0   = blockIdx.x * CHUNK;

    // A fragment source: mask row (m0+mr), 4 consecutive pixels -> b128 load
    const float* aptr = mask + (size_t)(m0 + mr) * P + 4 * half;

    // B fragment sources for column tiles 0..3 (col = 16t + mr)
    const float* bptr[4];
#pragma unroll
    for (int t = 0; t < 4; ++t) {
        const int col = 16 * t + mr;
        const int b   = col >> 3;
        const int c   = col & 7;
        const float* p;
        if (c < 4)      p = quat   + ((size_t)b * 4 + c) * P;
        else if (c < 7) p = scales + ((size_t)b * 3 + (c - 4)) * P;
        else            p = z      + (size_t)b * P;
        bptr[t] = p + 4 * half;
    }

    // Column tile 4: ones in column 64 only (constant B fragment, no traffic)
    v2f bones;
    bones.x = (mr == 0) ? 1.0f : 0.0f;
    bones.y = bones.x;

    v8f acc[5] = {};   // 5 independent accumulator chains -> WMMA ILP

#pragma unroll 2
    for (int k = k0; k < k0 + CHUNK; k += 8) {
        v4f a4 = *(const v4f*)(aptr + k);
        v2f aA; aA.x = a4.x; aA.y = a4.y;   // pixels {0,1,4,5} of this group
        v2f aB; aB.x = a4.z; aB.y = a4.w;   // pixels {2,3,6,7}

        acc[4] = __builtin_amdgcn_wmma_f32_16x16x4_f32(
                     false, aA, false, bones, (short)0, acc[4], false, false);
        acc[4] = __builtin_amdgcn_wmma_f32_16x16x4_f32(
                     false, aB, false, bones, (short)0, acc[4], false, false);
#pragma unroll
        for (int t = 0; t < 4; ++t) {
            v4f b4 = *(const v4f*)(bptr[t] + k);
            v2f bA; bA.x = b4.x; bA.y = b4.y;
            v2f bB; bB.x = b4.z; bB.y = b4.w;
            acc[t] = __builtin_amdgcn_wmma_f32_16x16x4_f32(
                         false, aA, false, bA, (short)0, acc[t], false, false);
            acc[t] = __builtin_amdgcn_wmma_f32_16x16x4_f32(
                         false, aB, false, bB, (short)0, acc[t], false, false);
        }
    }

    // Store deterministic per-block partial tile (fully overwritten each call)
    float* out = partials + (size_t)blockIdx.x * (NINST * NCOL);
#pragma unroll
    for (int t = 0; t < 5; ++t) {
        const int col = 16 * t + mr;
#pragma unroll
        for (int v = 0; v < 8; ++v) {
            const int row = m0 + v + 8 * half;   // C layout: VGPR v -> M = v + 8*half
            out[row * NCOL + col] = acc[t][v];
        }
    }
}

// ---------------------------------------------------------------------------
// Kernel 2: S[j] = sum_blk partials[blk][j]  (deterministic, coalesced over j)
// ---------------------------------------------------------------------------
__global__ __launch_bounds__(256)
void reduce_partials_kernel(const float* __restrict__ partials,
                            float* __restrict__ S)
{
    const int j = blockIdx.x * 256 + threadIdx.x;
    if (j >= NINST * NCOL) return;
    float s = 0.0f;
    for (int b = 0; b < GEMM_BLOCKS; ++b)
        s += partials[(size_t)b * (NINST * NCOL) + j];
    S[j] = s;
}

// ---------------------------------------------------------------------------
// Kernel 3: epilogue — select b == sample_ids[n], divide by mask_size,
// L2-normalize quaternion, exp(z).  Output layout (flat, return order):
//   q_agg [64,4] @0, s_agg [64,3] @256, z_agg [64,1] @448, xy @512
// ---------------------------------------------------------------------------
__global__ __launch_bounds__(64)
void finalize_kernel(const float* __restrict__ S,
                     const int* __restrict__ sids,
                     float* __restrict__ out)
{
    const int n = threadIdx.x;
    if (n >= NINST) return;
    const float inv = 1.0f / S[n * NCOL + 64];      // 1 / mask_size[n]
    const int sid   = sids[n];
    const float* row = S + n * NCOL + sid * 8;

    const float q0 = row[0] * inv, q1 = row[1] * inv;
    const float q2 = row[2] * inv, q3 = row[3] * inv;
    const float nrm = sqrtf(q0 * q0 + q1 * q1 + q2 * q2 + q3 * q3);
    const float qd  = 1.0f / fmaxf(nrm, 1e-12f);
    out[n * 4 + 0] = q0 * qd;
    out[n * 4 + 1] = q1 * qd;
    out[n * 4 + 2] = q2 * qd;
    out[n * 4 + 3] = q3 * qd;

    out[256 + n * 3 + 0] = row[4] * inv;
    out[256 + n * 3 + 1] = row[5] * inv;
    out[256 + n * 3 + 2] = row[6] * inv;

    out[448 + n] = expf(row[7] * inv);
}

// ---------------------------------------------------------------------------
// Kernel 4: xy_masked[n,c,p] = mask[n,p] * xy[sid[n],c,p]  (118 MB stream)
// float4 loads, non-temporal float4 stores (output has no reuse).
// ---------------------------------------------------------------------------
__global__ __launch_bounds__(256)
void xy_mask_kernel(const float* __restrict__ mask,
                    const int* __restrict__ sids,
                    const float* __restrict__ xy,
                    float* __restrict__ out)
{
    const unsigned P4 = P / 4;                       // 57600 float4 per plane
    const unsigned t  = blockIdx.x * 256u + threadIdx.x;  // < 128*P4
    const unsigned p4 = t % P4;
    const unsigned nc = t / P4;                      // 0..127
    const unsigned c  = nc & 1u;
    const unsigned n  = nc >> 1;
    const int sid = sids[n];

    v4f m = *(const v4f*)(mask + (size_t)n * P + (size_t)p4 * 4);
    v4f v = *(const v4f*)(xy + ((size_t)sid * 2 + c) * P + (size_t)p4 * 4);
    v4f r = m * v;
    __builtin_nontemporal_store(r, (v4f*)(out + (size_t)t * 4));
}

// ---------------------------------------------------------------------------
extern "C" void kernel_launch(void* const* d_in, const int* in_sizes, int n_in,
                              void* d_out, int out_size, void* d_ws, size_t ws_size,
                              hipStream_t stream)
{
    const float* mask   = (const float*)d_in[0];   // [64,480,480]
    const int*   sids   = (const int*)  d_in[1];   // [64]
    const float* quat   = (const float*)d_in[2];   // [8,4,480,480]
    const float* scales = (const float*)d_in[3];   // [8,3,480,480]
    const float* xy     = (const float*)d_in[4];   // [8,2,480,480]
    const float* z      = (const float*)d_in[5];   // [8,480,480]
    float* out = (float*)d_out;

    float* partials = (float*)d_ws;                          // 225*5120 floats
    float* S        = partials + (size_t)GEMM_BLOCKS * NINST * NCOL; // 5120 floats

    gemm_partial_kernel<<<GEMM_BLOCKS, 128, 0, stream>>>(mask, quat, scales, z, partials);
    reduce_partials_kernel<<<(NINST * NCOL + 255) / 256, 256, 0, stream>>>(partials, S);
    finalize_kernel<<<1, 64, 0, stream>>>(S, sids, out);

    const unsigned total4 = 128u * (P / 4);                  // 7,372,800 float4
    xy_mask_kernel<<<total4 / 256, 256, 0, stream>>>(mask, sids, xy, out + 512);
}